// SRCCLoss_42812234006687
// MI455X (gfx1250) — compile-verified
//
#include <hip/hip_runtime.h>
#include <math.h>

typedef __attribute__((ext_vector_type(2))) float        v2f;
typedef __attribute__((ext_vector_type(8))) float        v8f;
typedef __attribute__((ext_vector_type(2))) int          v2i;
typedef __attribute__((ext_vector_type(4))) unsigned int u32x4;
typedef __attribute__((ext_vector_type(4))) int          i32x4;
typedef __attribute__((ext_vector_type(8))) int          i32x8;

#define LOCAL_ELEMS   1024
#define LOCAL_THREADS 512
#define RB            256   // reduction blocks

// ------------------------------------------------------------------ TDM helper
// 1D tile (nelem 4-byte elements) DMA from global memory into LDS via the
// Tensor Data Mover.  D# group0: count=1 | lds_addr | global_addr | type=2.
// D# group1: data_size=4B, tensor_dim0=nelem, tensor_dim1=1, tile_dim0=nelem,
// tile_dim1=1, tensor_dim0_stride=nelem.  Groups 2/3 unused (<=2D tensor).
__device__ __forceinline__ void tdm_load_1d_b32(unsigned long long gaddr,
                                                unsigned int lds_off, int nelem) {
  u32x4 g0;
  g0[0] = 1u;                                               // count=1 (user mode)
  g0[1] = lds_off;                                          // lds_addr[31:0]
  g0[2] = (unsigned int)(gaddr & 0xFFFFFFFFull);            // global_addr[31:0]
  g0[3] = (unsigned int)((gaddr >> 32) & 0x01FFFFFFull)     // global_addr[56:32]
        | (2u << 30);                                       // type = 2 ("image")
  i32x8 g1;
  g1[0] = (int)(2u << 16);                                  // data_size = 4 bytes
  g1[1] = (int)(((unsigned)nelem & 0xFFFFu) << 16);         // tensor_dim0[15:0]
  g1[2] = (int)((((unsigned)nelem >> 16) & 0xFFFFu)         // tensor_dim0[31:16]
        | (1u << 16));                                      // tensor_dim1 = 1
  g1[3] = (int)(((unsigned)nelem & 0xFFFFu) << 16);         // tile_dim0
  g1[4] = 1;                                                // tile_dim1 = 1
  g1[5] = nelem;                                            // tensor_dim0_stride[31:0]
  g1[6] = 0;
  g1[7] = 0;
  i32x4 z4 = {0, 0, 0, 0};
#if defined(__clang_major__) && (__clang_major__ >= 23)
  i32x8 z8 = {0, 0, 0, 0, 0, 0, 0, 0};
  __builtin_amdgcn_tensor_load_to_lds(g0, g1, z4, z4, z8, 0);
#else
  __builtin_amdgcn_tensor_load_to_lds(g0, g1, z4, z4, 0);
#endif
}

// ---------------------------------------------------------------- init
__global__ void srcc_init_keys(const float* __restrict__ x, float* __restrict__ key,
                               int* __restrict__ idx, int n, float reg) {
  int i = blockIdx.x * blockDim.x + threadIdx.x;
  if (i < n) { key[i] = x[i] / reg; idx[i] = i; }
}

// ------------------------------------------------- bitonic sort (descending)
// direction convention: ((gidx & k) == 0) -> keep key[i] >= key[i+j] (descending run)

__global__ void srcc_bitonic_local_sort(float* __restrict__ key, int* __restrict__ idx) {
  __shared__ __align__(16) float sk[LOCAL_ELEMS];
  __shared__ __align__(16) int   si[LOCAL_ELEMS];
  const int tid  = threadIdx.x;
  const int base = blockIdx.x * LOCAL_ELEMS;
  // vectorized B64 tile load
  *(v2f*)&sk[2 * tid] = *(const v2f*)(key + base + 2 * tid);
  *(v2i*)&si[2 * tid] = *(const v2i*)(idx + base + 2 * tid);
  for (int k = 2; k <= LOCAL_ELEMS; k <<= 1) {
    for (int j = k >> 1; j > 0; j >>= 1) {
      __syncthreads();
      int i = ((tid & ~(j - 1)) << 1) | (tid & (j - 1));
      int l = i + j;
      bool desc = (((base + i) & k) == 0);
      float a = sk[i], b = sk[l];
      bool sw = desc ? (a < b) : (a > b);
      if (sw) { sk[i] = b; sk[l] = a; int t = si[i]; si[i] = si[l]; si[l] = t; }
    }
  }
  __syncthreads();
  *(v2f*)(key + base + 2 * tid) = *(const v2f*)&sk[2 * tid];
  *(v2i*)(idx + base + 2 * tid) = *(const v2i*)&si[2 * tid];
}

__global__ void srcc_bitonic_global_pass(float* __restrict__ key, int* __restrict__ idx,
                                         int j, int k) {
  int tid = blockIdx.x * blockDim.x + threadIdx.x;   // one thread per pair
  int i = ((tid & ~(j - 1)) << 1) | (tid & (j - 1));
  int l = i + j;
  bool desc = ((i & k) == 0);
  float a = key[i], b = key[l];
  bool sw = desc ? (a < b) : (a > b);
  if (sw) { key[i] = b; key[l] = a; int t = idx[i]; idx[i] = idx[l]; idx[l] = t; }
}

// all stages j = 512 .. 1 for a given k, inside LDS; tile staged via TDM DMA
__global__ void srcc_bitonic_local_tail(float* __restrict__ key, int* __restrict__ idx, int k) {
  __shared__ __align__(16) float sk[LOCAL_ELEMS];
  __shared__ __align__(16) int   si[LOCAL_ELEMS];
  const int tid  = threadIdx.x;
  const int base = blockIdx.x * LOCAL_ELEMS;
  if (tid == 0) {
    tdm_load_1d_b32((unsigned long long)(uintptr_t)(key + base),
                    (unsigned int)(uintptr_t)&sk[0], LOCAL_ELEMS);
    tdm_load_1d_b32((unsigned long long)(uintptr_t)(idx + base),
                    (unsigned int)(uintptr_t)&si[0], LOCAL_ELEMS);
    __builtin_amdgcn_s_wait_tensorcnt(0);   // issuing wave waits; barrier releases rest
  }
  __syncthreads();
  for (int j = LOCAL_ELEMS >> 1; j > 0; j >>= 1) {
    int i = ((tid & ~(j - 1)) << 1) | (tid & (j - 1));
    int l = i + j;
    bool desc = (((base + i) & k) == 0);
    float a = sk[i], b = sk[l];
    bool sw = desc ? (a < b) : (a > b);
    if (sw) { sk[i] = b; sk[l] = a; int t = si[i]; si[i] = si[l]; si[l] = t; }
    __syncthreads();
  }
  *(v2f*)(key + base + 2 * tid) = *(const v2f*)&sk[2 * tid];
  *(v2i*)(idx + base + 2 * tid) = *(const v2i*)&si[2 * tid];
}

// -------------------------------------------------- PAV (sequential, exact)
// y[i] = s[i] - (n - i); non-increasing isotonic fit via pool-adjacent-violators.
__global__ void srcc_pav(const float* __restrict__ s, float* __restrict__ means,
                         float* __restrict__ counts, int* __restrict__ pstart,
                         int* __restrict__ npools, int n) {
  if (threadIdx.x != 0 || blockIdx.x != 0) return;
  int top = 0;
  for (int i = 0; i < n; ++i) {
    float y = s[i] - (float)(n - i);
    means[top]  = y;
    counts[top] = 1.0f;
    ++top;
    while (top > 1 && means[top - 1] > means[top - 2]) {
      float ct = counts[top - 1] + counts[top - 2];
      float mt = (means[top - 1] * counts[top - 1] + means[top - 2] * counts[top - 2]) / ct;
      means[top - 2]  = mt;
      counts[top - 2] = ct;
      --top;
    }
  }
  int off = 0;
  for (int p = 0; p < top; ++p) { pstart[p] = off; off += (int)counts[p]; }
  *npools = top;
}

// ------------------------------------- expand pooled means + undo permutation
__global__ void srcc_expand_scatter(const float* __restrict__ s, const int* __restrict__ idx,
                                    const float* __restrict__ means, const int* __restrict__ pstart,
                                    const int* __restrict__ npools, float* __restrict__ ranks, int n) {
  int i = blockIdx.x * blockDim.x + threadIdx.x;
  if (i >= n) return;
  int np = *npools;
  int lo = 0, hi = np - 1;                // largest p with pstart[p] <= i
  while (lo < hi) {
    int mid = (lo + hi + 1) >> 1;
    if (pstart[mid] <= i) lo = mid; else hi = mid - 1;
  }
  float primal = s[i] - means[lo];
  ranks[idx[i]] = primal;
}

// -------------------------------------------------------- reductions (pass 1)
__global__ void srcc_sum_pass(const float* __restrict__ ra, const float* __restrict__ rb,
                              float* __restrict__ partials, int n) {
  __shared__ float sa[256], sb[256];
  int tid = threadIdx.x;
  int gid = blockIdx.x * blockDim.x + tid;
  int stride = gridDim.x * blockDim.x;
  float suma = 0.f, sumb = 0.f;
  for (int i = gid; i < n; i += stride) { suma += ra[i]; sumb += rb[i]; }
  sa[tid] = suma; sb[tid] = sumb;
  __syncthreads();
  for (int s = 128; s > 0; s >>= 1) {
    if (tid < s) { sa[tid] += sa[tid + s]; sb[tid] += sb[tid + s]; }
    __syncthreads();
  }
  if (tid == 0) { partials[blockIdx.x] = sa[0]; partials[gridDim.x + blockIdx.x] = sb[0]; }
}

__global__ void srcc_reduce_means(const float* __restrict__ partials, int nb,
                                  float* __restrict__ meanbuf, float inv_n) {
  if (threadIdx.x != 0 || blockIdx.x != 0) return;
  float a = 0.f, b = 0.f;
  for (int i = 0; i < nb; ++i) { a += partials[i]; b += partials[nb + i]; }
  meanbuf[0] = a * inv_n;
  meanbuf[1] = b * inv_n;
}

// ---------------------------------------- pass 2: centered second moments, WMMA
// One 64-element chunk per wave iteration. For V_WMMA_F32_16X16X4_F32 the A(16x4)
// and B(4x16) register layouts coincide (lane = khi*16 + m/n, vgpr = k&1), so one
// fragment built as frag[lane] = {x[base+4*(lane&15)+koff], x[...+1]} serves both
// roles and the accumulator diagonal D[m][m] = sum_k x[4m+k]*y[4m+k].
// Chunk loop is driven by readfirstlane'd scalars so it lowers to s_cmp/s_cbranch
// and EXEC provably stays all-ones around the WMMAs (ISA requirement).
__global__ void srcc_corr_wmma(const float* __restrict__ ra, const float* __restrict__ rb,
                               const float* __restrict__ meanbuf, float* __restrict__ partials,
                               int n, int nbp) {
  const int lane           = threadIdx.x & 31;
  const int wave_in_block  = threadIdx.x >> 5;
  const int waves_per_blk  = blockDim.x >> 5;
  const int wave_id        = __builtin_amdgcn_readfirstlane(blockIdx.x * waves_per_blk + wave_in_block);
  const int total_waves    = gridDim.x * waves_per_blk;
  const float ma = meanbuf[0];
  const float mb = meanbuf[1];

  const int m    = lane & 15;
  const int koff = (lane >> 4) << 1;       // lanes 0-15 -> K{0,1}; lanes 16-31 -> K{2,3}

  float acc_xx = 0.f, acc_yy = 0.f, acc_xy = 0.f;

  const int nchunks = n >> 6;
  for (int c = wave_id; c < nchunks; c += total_waves) {   // scalar loop: EXEC untouched
    const int base = c << 6;
    const int off  = base + (m << 2) + koff;
    v2f fa, fb;
    fa.x = ra[off] - ma;  fa.y = ra[off + 1] - ma;
    fb.x = rb[off] - mb;  fb.y = rb[off + 1] - mb;
    v8f cxx = {}; v8f cyy = {}; v8f cxy = {};
    cxx = __builtin_amdgcn_wmma_f32_16x16x4_f32(false, fa, false, fa, (short)0, cxx, false, false);
    cyy = __builtin_amdgcn_wmma_f32_16x16x4_f32(false, fb, false, fb, (short)0, cyy, false, false);
    cxy = __builtin_amdgcn_wmma_f32_16x16x4_f32(false, fa, false, fb, (short)0, cxy, false, false);
    // diagonal: VGPR v holds D[v][lane] (lanes 0-15) and D[v+8][lane-16] (lanes 16-31)
#pragma unroll
    for (int v = 0; v < 8; ++v) {
      const bool on_diag = (lane == v) || (lane == v + 24);
      acc_xx += on_diag ? cxx[v] : 0.f;
      acc_yy += on_diag ? cyy[v] : 0.f;
      acc_xy += on_diag ? cxy[v] : 0.f;
    }
  }
  for (int s = 16; s > 0; s >>= 1) {          // wave32 shuffle reduce
    acc_xx += __shfl_xor(acc_xx, s, 32);
    acc_yy += __shfl_xor(acc_yy, s, 32);
    acc_xy += __shfl_xor(acc_xy, s, 32);
  }
  __shared__ float wxx[32], wyy[32], wxy[32];
  if (lane == 0) { wxx[wave_in_block] = acc_xx; wyy[wave_in_block] = acc_yy; wxy[wave_in_block] = acc_xy; }
  __syncthreads();
  if (threadIdx.x == 0) {
    float txx = 0.f, tyy = 0.f, txy = 0.f;
    for (int w = 0; w < waves_per_blk; ++w) { txx += wxx[w]; tyy += wyy[w]; txy += wxy[w]; }
    partials[blockIdx.x]           = txy;
    partials[nbp + blockIdx.x]     = txx;
    partials[2 * nbp + blockIdx.x] = tyy;
  }
}

__global__ void srcc_final_loss(const float* __restrict__ p2, int nb, float* __restrict__ out) {
  if (threadIdx.x != 0 || blockIdx.x != 0) return;
  double sxy = 0.0, sxx = 0.0, syy = 0.0;
  for (int i = 0; i < nb; ++i) {
    sxy += (double)p2[i];
    sxx += (double)p2[nb + i];
    syy += (double)p2[2 * nb + i];
  }
  out[0] = (float)(sxy / (sqrt(sxx) * sqrt(syy)));
}

// ----------------------------------------------------------------- launcher
extern "C" void kernel_launch(void* const* d_in, const int* in_sizes, int n_in,
                              void* d_out, int out_size, void* d_ws, size_t ws_size,
                              hipStream_t stream) {
  const float* input  = (const float*)d_in[0];
  const float* target = (const float*)d_in[1];
  float* out = (float*)d_out;
  const int n = in_sizes[0];
  const size_t N = (size_t)n;

  float* ws      = (float*)d_ws;
  float* keyA    = ws;                 int* idxA    = (int*)(ws + N);
  float* keyB    = ws + 2 * N;         int* idxB    = (int*)(ws + 3 * N);
  float* meansA  = ws + 4 * N;         float* countsA = ws + 5 * N;
  float* meansB  = ws + 6 * N;         float* countsB = ws + 7 * N;
  int*   pstartA = (int*)(ws + 8 * N);
  int*   pstartB = (int*)(ws + 9 * N);
  float* ranksA  = ws + 10 * N;        float* ranksB  = ws + 11 * N;
  float* scratch = ws + 12 * N;
  float* p1      = scratch;            // 2*RB
  float* meanbuf = scratch + 2 * RB;   // 2
  float* p2      = meanbuf + 2;        // 3*RB
  int*   npA     = (int*)(p2 + 3 * RB);
  int*   npB     = npA + 1;

  const float reg = 0.1f;
  const int ib = (n + 255) / 256;
  srcc_init_keys<<<ib, 256, 0, stream>>>(input,  keyA, idxA, n, reg);
  srcc_init_keys<<<ib, 256, 0, stream>>>(target, keyB, idxB, n, reg);

  const int lblocks = n / LOCAL_ELEMS;
  const int gblocks = (n / 2) / 256;
  auto sort_array = [&](float* key, int* idx) {
    srcc_bitonic_local_sort<<<lblocks, LOCAL_THREADS, 0, stream>>>(key, idx);
    for (int k = 2 * LOCAL_ELEMS; k <= n; k <<= 1) {
      for (int j = k >> 1; j >= LOCAL_ELEMS; j >>= 1)
        srcc_bitonic_global_pass<<<gblocks, 256, 0, stream>>>(key, idx, j, k);
      srcc_bitonic_local_tail<<<lblocks, LOCAL_THREADS, 0, stream>>>(key, idx, k);
    }
  };
  sort_array(keyA, idxA);
  sort_array(keyB, idxB);

  srcc_pav<<<1, 32, 0, stream>>>(keyA, meansA, countsA, pstartA, npA, n);
  srcc_pav<<<1, 32, 0, stream>>>(keyB, meansB, countsB, pstartB, npB, n);

  srcc_expand_scatter<<<ib, 256, 0, stream>>>(keyA, idxA, meansA, pstartA, npA, ranksA, n);
  srcc_expand_scatter<<<ib, 256, 0, stream>>>(keyB, idxB, meansB, pstartB, npB, ranksB, n);

  srcc_sum_pass<<<RB, 256, 0, stream>>>(ranksA, ranksB, p1, n);
  srcc_reduce_means<<<1, 32, 0, stream>>>(p1, RB, meanbuf, 1.0f / (float)n);

  srcc_corr_wmma<<<RB, 256, 0, stream>>>(ranksA, ranksB, meanbuf, p2, n, RB);
  srcc_final_loss<<<1, 32, 0, stream>>>(p2, RB, out);
}